// Generator_79989470921100
// MI455X (gfx1250) — compile-verified
//
#include <hip/hip_runtime.h>
#include <hip/hip_bf16.h>

typedef float v2f __attribute__((ext_vector_type(2)));
typedef float v8f __attribute__((ext_vector_type(8)));

#define DIM_IN 128
#define H1 1000
#define H2 100
#define H3 16
#define DTGT 16
#define NT 4   // 16-col tiles per wave (16x64 C strip, A frag reused 4x)

// ---------------------------------------------------------------------------
// GEMM  C[M,Nc] = A[M,K] * B[K,Nc]   (fp32, V_WMMA_F32_16X16X4_F32)
// One wave32 computes a 16x(16*NT) strip of C. Requires M%16==0, K%4==0, K even.
// Per ISA 7.12.2: A frag lane l: row m=l&15, a=(A[m][k+2h], A[m][k+2h+1]), h=l>>4
//                 B frag lane l: col n+(l&15), rows k+2h, k+2h+1
//                 C/D: VGPR v -> row m0+v+8h, col n+(l&15)
// Out-of-range columns are clamped to Nc-1 for loads (valid memory, garbage
// result) and masked only at the final store, so EXEC is full at every WMMA
// and the inner loop is branch-free. B is addressed as wave-uniform base
// (scalar-bumped by 4*Nc per k-step) + small per-lane int offset so the
// backend can use saddr-form global loads (no per-load 64-bit VALU adds).
// ---------------------------------------------------------------------------
__global__ void gemm_wmma_f32(const float* __restrict__ A, const float* __restrict__ B,
                              float* __restrict__ C, int M, int K, int Nc) {
  const int ntiles  = (Nc + 15) >> 4;
  const int nstrips = (ntiles + NT - 1) / NT;
  const int mtiles  = M >> 4;
  const int wave = (int)(blockIdx.x * (blockDim.x >> 5) + (threadIdx.x >> 5));
  if (wave >= mtiles * nstrips) return;       // wave-uniform exit
  const int mt   = wave / nstrips;
  const int st   = wave % nstrips;
  const int lane = threadIdx.x & 31;
  const int half = lane >> 4;
  const int l16  = lane & 15;
  const int m0   = mt << 4;
  const int n0   = (st * NT) << 4;

  int  boff[NT];                               // per-lane int offset into B row block
  int  col[NT];
  bool cv[NT];
#pragma unroll
  for (int t = 0; t < NT; ++t) {
    int c  = n0 + t * 16 + l16;
    cv[t]  = (c < Nc);
    col[t] = cv[t] ? c : (Nc - 1);            // clamp: loads stay in-bounds
    boff[t] = 2 * half * Nc + col[t];         // row (k+2h), col — rel. to k-block base
  }

  const float* pA = A + (size_t)(m0 + l16) * K + 2 * half;  // per-lane A row
  const float* pB = B;                                       // wave-uniform k-block base
  v8f acc[NT] = {};

  for (int k = 0; k < K; k += 4) {
    v2f a = *(const v2f*)pA;                  // 8B-aligned: K even, k%4==0
    pA += 4;
#pragma unroll
    for (int t = 0; t < NT; ++t) {
      v2f b;
      b.x = pB[boff[t]];                      // row k+2h
      b.y = pB[boff[t] + Nc];                 // row k+2h+1
      acc[t] = __builtin_amdgcn_wmma_f32_16x16x4_f32(false, a, false, b, (short)0,
                                                     acc[t], false, false);
    }
    pB += (size_t)4 * Nc;                     // uniform scalar bump
  }

#pragma unroll
  for (int t = 0; t < NT; ++t) {
    if (cv[t]) {
      float* crow = C + (size_t)(m0 + 8 * half) * Nc + n0 + t * 16 + l16;
#pragma unroll
      for (int v = 0; v < 8; ++v) crow[(size_t)v * Nc] = acc[t][v];
    }
  }
}

// ---------------------------------------------------------------------------
// Degree / normalization helpers
// ---------------------------------------------------------------------------
__global__ void k_fill(float* __restrict__ p, float v, int n) {
  int i = blockIdx.x * blockDim.x + threadIdx.x;
  if (i < n) p[i] = v;
}

__global__ void k_deg(const int* __restrict__ dst, float* __restrict__ deg, int E) {
  int e = blockIdx.x * blockDim.x + threadIdx.x;
  if (e < E) atomicAdd(&deg[dst[e]], 1.0f);
}

__global__ void k_rsqrt(float* __restrict__ p, int n) {
  int i = blockIdx.x * blockDim.x + threadIdx.x;
  if (i < n) {
    float d = p[i];
    p[i] = (d > 0.0f) ? rsqrtf(d) : 0.0f;
  }
}

// out[n][f] = bias[f] + h[n][f] * dinv[n]^2   (self-loop term; single writer)
__global__ void k_init_self(const float* __restrict__ h, const float* __restrict__ dinv,
                            const float* __restrict__ bias, float* __restrict__ out,
                            int total, int F) {
  int i = blockIdx.x * blockDim.x + threadIdx.x;
  if (i >= total) return;
  int n = i / F, f = i % F;
  float di = dinv[n];
  out[i] = bias[f] + h[i] * (di * di);
}

// out[dst[e]][4g..4g+3] += h[src[e]][4g..4g+3] * dinv[s]*dinv[d]
// thread = (edge, float4 group); consecutive threads -> coalesced row access,
// atomic destinations are L2-resident (<=80MB vs 192MB L2).
__global__ void k_edges(const float* __restrict__ h, const int* __restrict__ src,
                        const int* __restrict__ dst, const float* __restrict__ dinv,
                        float* __restrict__ out, int E, int F4) {
  long long i = (long long)blockIdx.x * blockDim.x + threadIdx.x;
  long long total = (long long)E * F4;
  if (i >= total) return;
  int e = (int)(i / F4);
  int g = (int)(i % F4);
  int s = src[e], d = dst[e];
  float nm = dinv[s] * dinv[d];
  const float4 v = *(const float4*)(h + (size_t)s * F4 * 4 + g * 4);
  float* o = out + (size_t)d * F4 * 4 + g * 4;
  atomicAdd(o + 0, v.x * nm);
  atomicAdd(o + 1, v.y * nm);
  atomicAdd(o + 2, v.z * nm);
  atomicAdd(o + 3, v.w * nm);
}

// pred = sigmoid(z + bl[f]); written to both pred_s and pred_t (reference bug
// uses hs for both predictions, kept faithfully).
__global__ void k_sigmoid(const float* __restrict__ z, const float* __restrict__ bl,
                          float* __restrict__ ps, float* __restrict__ pt,
                          int total, int F) {
  int i = blockIdx.x * blockDim.x + threadIdx.x;
  if (i >= total) return;
  float v = z[i] + bl[i % F];
  v = 1.0f / (1.0f + __expf(-v));
  ps[i] = v;
  pt[i] = v;
}

// ---------------------------------------------------------------------------
static inline int cdiv(long long a, long long b) { return (int)((a + b - 1) / b); }

extern "C" void kernel_launch(void* const* d_in, const int* in_sizes, int n_in,
                              void* d_out, int out_size, void* d_ws, size_t ws_size,
                              hipStream_t stream) {
  const float* x_s  = (const float*)d_in[0];
  const int*   ei_s = (const int*)  d_in[1];
  const float* x_t  = (const float*)d_in[2];
  const int*   ei_t = (const int*)  d_in[3];
  const float* W1 = (const float*)d_in[4];  const float* b1 = (const float*)d_in[5];
  const float* W2 = (const float*)d_in[6];  const float* b2 = (const float*)d_in[7];
  const float* W3 = (const float*)d_in[8];  const float* b3 = (const float*)d_in[9];
  const float* Wl = (const float*)d_in[10]; const float* bl = (const float*)d_in[11];

  const int N = in_sizes[0] / DIM_IN;   // 20000 (multiple of 16)
  const int E = in_sizes[1] / 2;        // 320000

  float* out  = (float*)d_out;          // [hs | ht | pred_s | pred_t], each N*16
  float* bufA = (float*)d_ws;           // N*H1 floats
  float* bufB = bufA + (size_t)N * H1;  // N*H1 floats
  float* dinv = bufB + (size_t)N * H1;  // N floats

  const int BS  = 256;                  // 8 waves / block
  const int WPB = BS / 32;
  const int mtiles = N / 16;

  auto gemm_grid = [&](int Nc) {
    int strips = (cdiv(Nc, 16) + NT - 1) / NT;
    return cdiv((long long)mtiles * strips, WPB);
  };

  for (int g = 0; g < 2; ++g) {
    const float* x   = g ? x_t : x_s;
    const int*   src = g ? ei_t : ei_s;
    const int*   dst = src + E;
    float* hout = out + (size_t)g * N * H3;

    // D^-1/2 with self-loops: deg = 1 + indegree, then rsqrt in place
    k_fill <<<cdiv(N, BS), BS, 0, stream>>>(dinv, 1.0f, N);
    k_deg  <<<cdiv(E, BS), BS, 0, stream>>>(dst, dinv, E);
    k_rsqrt<<<cdiv(N, BS), BS, 0, stream>>>(dinv, N);

    // Layer 1: [N,128] x [128,1000]
    gemm_wmma_f32<<<gemm_grid(H1), BS, 0, stream>>>(x, W1, bufA, N, DIM_IN, H1);
    k_init_self<<<cdiv((long long)N * H1, BS), BS, 0, stream>>>(bufA, dinv, b1, bufB,
                                                                N * H1, H1);
    k_edges<<<cdiv((long long)E * (H1 / 4), BS), BS, 0, stream>>>(bufA, src, dst, dinv,
                                                                  bufB, E, H1 / 4);

    // Layer 2: [N,1000] x [1000,100]
    gemm_wmma_f32<<<gemm_grid(H2), BS, 0, stream>>>(bufB, W2, bufA, N, H1, H2);
    k_init_self<<<cdiv((long long)N * H2, BS), BS, 0, stream>>>(bufA, dinv, b2, bufB,
                                                                N * H2, H2);
    k_edges<<<cdiv((long long)E * (H2 / 4), BS), BS, 0, stream>>>(bufA, src, dst, dinv,
                                                                  bufB, E, H2 / 4);

    // Layer 3: [N,100] x [100,16] -> h{s,t} directly in d_out
    gemm_wmma_f32<<<gemm_grid(H3), BS, 0, stream>>>(bufB, W3, bufA, N, H2, H3);
    k_init_self<<<cdiv((long long)N * H3, BS), BS, 0, stream>>>(bufA, dinv, b3, hout,
                                                                N * H3, H3);
    k_edges<<<cdiv((long long)E * (H3 / 4), BS), BS, 0, stream>>>(bufA, src, dst, dinv,
                                                                  hout, E, H3 / 4);
  }

  // pred = sigmoid(hs @ Wl + bl); reference uses hs for BOTH preds.
  const float* hs = out;
  gemm_wmma_f32<<<gemm_grid(DTGT), BS, 0, stream>>>(hs, Wl, bufA, N, H3, DTGT);
  k_sigmoid<<<cdiv((long long)N * DTGT, BS), BS, 0, stream>>>(
      bufA, bl, out + (size_t)2 * N * DTGT, out + (size_t)3 * N * DTGT, N * DTGT, DTGT);
}